// AdderBase_75265006895563
// MI455X (gfx1250) — compile-verified
//
#include <hip/hip_runtime.h>
#include <stdint.h>

typedef float v2f  __attribute__((ext_vector_type(2)));
typedef float v8f  __attribute__((ext_vector_type(8)));
typedef unsigned int u32x4 __attribute__((ext_vector_type(4)));
typedef int   i32x4 __attribute__((ext_vector_type(4)));
typedef int   i32x8 __attribute__((ext_vector_type(8)));

#define CIN   512
#define COUT  512
#define ROWS  16          // rows per workgroup tile
#define KC    32          // k-chunk staged per double-buffer step
#define NCHUNK (CIN / KC) // 16
#define XPAD  20          // padded row stride (floats) of transposed x tile [KC][XPAD]
#define PSTR  66          // padded stride of 16x64 partial-sum matrix (8B aligned, bank-spread)

// ---- Tensor Data Mover: async copy of a KC x COUT f32 tile of w into LDS ----
// D# group0: count=1 | lds_addr | global_addr[56:0] | type=2
// D# group1: data_size=4B | tensor_dim0=COUT | tensor_dim1=KC | tile_dim0=COUT |
//            tile_dim1=KC | tensor_dim0_stride=COUT
__device__ __forceinline__ void tdm_load_w_chunk(const float* wsrc, unsigned lds_byte, int krow0)
{
  unsigned long long ga = (unsigned long long)(uintptr_t)(wsrc + (size_t)krow0 * COUT);
  u32x4 g0;
  g0[0] = 1u;                                                  // count=1, user mode
  g0[1] = lds_byte;                                            // LDS byte address
  g0[2] = (unsigned)(ga & 0xFFFFFFFFull);                      // global_addr[31:0]
  g0[3] = (unsigned)((ga >> 32) & 0x01FFFFFFull) | (2u << 30); // global_addr[56:32] | type=2
  i32x8 g1;
  g1[0] = (int)(2u << 16);                 // wg_mask=0, data_size=2 (4 bytes)
  g1[1] = (int)((unsigned)COUT << 16);     // tensor_dim0[15:0] << 16
  g1[2] = (int)((unsigned)KC   << 16);     // tensor_dim1[15:0] << 16 | tensor_dim0[31:16]=0
  g1[3] = (int)((unsigned)COUT << 16);     // tile_dim0 << 16   | tensor_dim1[31:16]=0
  g1[4] = (int)(unsigned)KC;               // tile_dim2=0 << 16 | tile_dim1=KC
  g1[5] = (int)(unsigned)COUT;             // tensor_dim0_stride[31:0]
  g1[6] = 0;                               // dim1_stride lo16 | dim0_stride hi16
  g1[7] = 0;
  i32x4 z4 = {0, 0, 0, 0};
#if defined(__clang_major__) && (__clang_major__ >= 23)
  i32x8 z8 = {0, 0, 0, 0, 0, 0, 0, 0};
  __builtin_amdgcn_tensor_load_to_lds(g0, g1, z4, z4, z8, 0);
#else
  __builtin_amdgcn_tensor_load_to_lds(g0, g1, z4, z4, 0);
#endif
}

// ---- exact f32 row reduction of a 16x64 LDS matrix via chained 16x16x4 f32 WMMA
// against a ones B-matrix. D column 0 holds the 16 row sums (lanes 0/16, VGPRs 0..7).
__device__ __forceinline__ v8f wmma_rowsum16x64(const float* Pb, int tid)
{
  const int m    = tid & 15;            // A-matrix M = lane%16
  const int koff = (tid & 16) ? 2 : 0;  // lanes 16..31 hold K=2,3
  v2f ones; ones[0] = 1.0f; ones[1] = 1.0f;
  v8f c = {0.0f, 0.0f, 0.0f, 0.0f, 0.0f, 0.0f, 0.0f, 0.0f};
#pragma unroll
  for (int kk = 0; kk < 16; ++kk) {
    v2f a = *(const v2f*)(Pb + m * PSTR + (kk << 2) + koff);
    c = __builtin_amdgcn_wmma_f32_16x16x4_f32(false, a, false, ones, (short)0, c, false, false);
  }
  return c;
}

__global__ __launch_bounds__(256) void adder_ln_kernel(
    const float* __restrict__ x,   const float* __restrict__ w,
    const float* __restrict__ mbi, const float* __restrict__ mbo,
    const float* __restrict__ gma, const float* __restrict__ bta,
    float* __restrict__ out)
{
  __shared__ float wLds[2][KC * COUT];   // 2 x 64 KB, TDM double buffer
  __shared__ float xT[2][KC * XPAD];     // transposed x tile (k-major), padded
  __shared__ float P[ROWS * PSTR + 8];   // 16x64 partial sums for LN reduction
  __shared__ float stats[32];            // [0..15]=mean, [16..31]=rstd

  const int tid  = threadIdx.x;
  const int tx   = tid & 63;   // 64 column-groups of 8
  const int ty   = tid >> 6;   // 4 row-groups of 4
  const int row0 = blockIdx.x * ROWS;
  const int c0   = tx * 8;

  // scalar (SGPR) wave-0 test: TDM ignores EXEC and WMMA requires EXEC all-ones,
  // so gate with a uniform branch, never an EXEC mask.
  const bool lead = (__builtin_amdgcn_readfirstlane((int)tid) == 0);

  // ---- prologue: stage chunk 0 ----
  if (lead) tdm_load_w_chunk(w, (unsigned)((uintptr_t)&wLds[0][0]), 0);
  if (tid < 128) {                       // 16 rows x 8 quads of 4 k-values
    const int r  = tid >> 3;
    const int k4 = tid & 7;
    const float4 xq = *(const float4*)&x[(size_t)(row0 + r) * CIN + 4 * k4];
    const float4 mq = *(const float4*)&mbi[4 * k4];
    float* p = &xT[0][(k4 << 2) * XPAD + r];
    p[0 * XPAD] = xq.x * mq.x;
    p[1 * XPAD] = xq.y * mq.y;
    p[2 * XPAD] = xq.z * mq.z;
    p[3 * XPAD] = xq.w * mq.w;
  }
  if (lead) __builtin_amdgcn_s_wait_tensorcnt(0);
  __syncthreads();

  float acc[4][8];
#pragma unroll
  for (int ri = 0; ri < 4; ++ri)
#pragma unroll
    for (int ci = 0; ci < 8; ++ci) acc[ri][ci] = 0.0f;

  // ---- main loop: double-buffered TDM w-stream + register-pipelined x-stream ----
  for (int c = 0; c < NCHUNK; ++c) {
    const int cur = c & 1, nxt = cur ^ 1;
    float4 xq, mq;
    const bool more = (c + 1 < NCHUNK);
    if (more) {
      if (tid < 128) {
        const int r  = tid >> 3;
        const int k4 = tid & 7;
        xq = *(const float4*)&x[(size_t)(row0 + r) * CIN + (c + 1) * KC + 4 * k4];
        mq = *(const float4*)&mbi[(c + 1) * KC + 4 * k4];
      }
      if (lead) tdm_load_w_chunk(w, (unsigned)((uintptr_t)&wLds[nxt][0]), (c + 1) * KC);
    }

    const float* wp = &wLds[cur][0];
    const float* xp = &xT[cur][0];
#pragma unroll 4
    for (int k = 0; k < KC; ++k) {
      const float4 xv = *(const float4*)(xp + k * XPAD + (ty << 2)); // LDS broadcast
      const float4 wa = *(const float4*)(wp + k * COUT + c0);
      const float4 wb = *(const float4*)(wp + k * COUT + c0 + 4);
      const float xr[4] = {xv.x, xv.y, xv.z, xv.w};
      const float wr[8] = {wa.x, wa.y, wa.z, wa.w, wb.x, wb.y, wb.z, wb.w};
#pragma unroll
      for (int ri = 0; ri < 4; ++ri)
#pragma unroll
        for (int ci = 0; ci < 8; ++ci)
          acc[ri][ci] += __builtin_fabsf(xr[ri] - wr[ci]); // v_sub + v_add |src|
    }

    if (more) {
      if (tid < 128) {
        const int r  = tid >> 3;
        const int k4 = tid & 7;
        float* p = &xT[nxt][(k4 << 2) * XPAD + r];
        p[0 * XPAD] = xq.x * mq.x;
        p[1 * XPAD] = xq.y * mq.y;
        p[2 * XPAD] = xq.z * mq.z;
        p[3 * XPAD] = xq.w * mq.w;
      }
      if (lead) __builtin_amdgcn_s_wait_tensorcnt(0);
    }
    __syncthreads();
  }

  // ---- epilogue: o = -acc * mbo, then fused LayerNorm over Cout=512 ----
  const float4 ma = *(const float4*)&mbo[c0];
  const float4 mb = *(const float4*)&mbo[c0 + 4];
  const float mo[8] = {ma.x, ma.y, ma.z, ma.w, mb.x, mb.y, mb.z, mb.w};
  float o[4][8];
#pragma unroll
  for (int ri = 0; ri < 4; ++ri)
#pragma unroll
    for (int ci = 0; ci < 8; ++ci) o[ri][ci] = -acc[ri][ci] * mo[ci];

  // pass 1: mean
#pragma unroll
  for (int ri = 0; ri < 4; ++ri) {
    float s = 0.0f;
#pragma unroll
    for (int ci = 0; ci < 8; ++ci) s += o[ri][ci];
    P[((ty << 2) + ri) * PSTR + tx] = s;
  }
  __syncthreads();
  if (lead) {
    v8f s = wmma_rowsum16x64(P, tid);
    if ((tid & 15) == 0) {
      const int base = (tid >> 4) << 3; // lane 0 -> rows 0..7, lane 16 -> rows 8..15
#pragma unroll
      for (int j = 0; j < 8; ++j) stats[base + j] = s[j] * (1.0f / (float)COUT);
    }
  }
  __syncthreads();

  // pass 2: centered variance (matches jnp.var numerically)
  float mrow[4];
#pragma unroll
  for (int ri = 0; ri < 4; ++ri) mrow[ri] = stats[(ty << 2) + ri];
#pragma unroll
  for (int ri = 0; ri < 4; ++ri) {
    float s = 0.0f;
#pragma unroll
    for (int ci = 0; ci < 8; ++ci) {
      const float d = o[ri][ci] - mrow[ri];
      s += d * d;
    }
    P[((ty << 2) + ri) * PSTR + tx] = s;
  }
  __syncthreads();
  if (lead) {
    v8f s = wmma_rowsum16x64(P, tid);
    if ((tid & 15) == 0) {
      const int base = (tid >> 4) << 3;
#pragma unroll
      for (int j = 0; j < 8; ++j) {
        const float var = s[j] * (1.0f / (float)COUT);
        stats[16 + base + j] = 1.0f / sqrtf(var + 1e-5f);
      }
    }
  }
  __syncthreads();

  const float4 ga = *(const float4*)&gma[c0];
  const float4 gb = *(const float4*)&gma[c0 + 4];
  const float4 ba = *(const float4*)&bta[c0];
  const float4 bb = *(const float4*)&bta[c0 + 4];
  const float gv[8] = {ga.x, ga.y, ga.z, ga.w, gb.x, gb.y, gb.z, gb.w};
  const float bv[8] = {ba.x, ba.y, ba.z, ba.w, bb.x, bb.y, bb.z, bb.w};
#pragma unroll
  for (int ri = 0; ri < 4; ++ri) {
    const int rr = (ty << 2) + ri;
    const float mn = mrow[ri];
    const float rs = stats[16 + rr];
    float y[8];
#pragma unroll
    for (int ci = 0; ci < 8; ++ci) y[ci] = (o[ri][ci] - mn) * rs * gv[ci] + bv[ci];
    float* op = &out[(size_t)(row0 + rr) * COUT + c0];
    *(float4*)(op)     = make_float4(y[0], y[1], y[2], y[3]);
    *(float4*)(op + 4) = make_float4(y[4], y[5], y[6], y[7]);
  }
}

extern "C" void kernel_launch(void* const* d_in, const int* in_sizes, int n_in,
                              void* d_out, int out_size, void* d_ws, size_t ws_size,
                              hipStream_t stream) {
  (void)n_in; (void)out_size; (void)d_ws; (void)ws_size;
  const float* x   = (const float*)d_in[0];
  const float* w   = (const float*)d_in[1];
  const float* mbi = (const float*)d_in[2];
  const float* mbo = (const float*)d_in[3];
  const float* g   = (const float*)d_in[4];
  const float* b   = (const float*)d_in[5];
  float* out = (float*)d_out;
  const int N = in_sizes[0] / CIN;          // 4096 token rows
  adder_ln_kernel<<<dim3(N / ROWS), dim3(256), 0, stream>>>(x, w, mbi, mbo, g, b, out);
}